// GeomGCN_layer_89361089560784
// MI455X (gfx1250) — compile-verified
//
#include <hip/hip_runtime.h>

typedef __attribute__((ext_vector_type(2))) float v2f;
typedef __attribute__((ext_vector_type(8))) float v8f;

#define N_NODES   50000
#define N_EDGES   800000
#define IN_F      256
#define OUT_F     64
#define NUM_DIV   9
#define NUM_HEADS 2
#define HD        (NUM_HEADS * NUM_DIV)   // 18
#define OUT_COLS  (HD * OUT_F)            // 1152
#define LDS_STRIDE 260                    // 256 + 4 pad -> conflict-free strided reads

// -----------------------------------------------------------------------------
// Kernel A: h_ws[hd][n][o] = sum_k (features[n][k]*norm[n]) * W[hd][o][k]
// One wave computes a 16(node) x 64(out) tile via 4 WMMA accumulators, K=256
// in steps of 4 using V_WMMA_F32_16X16X4_F32 (full fp32 precision).
// -----------------------------------------------------------------------------
__global__ __launch_bounds__(256) void geomgcn_gemm(
    const float* __restrict__ features,
    const float* __restrict__ norm,
    const float* __restrict__ W,
    float* __restrict__ h_ws) {
  __shared__ float w_lds[OUT_F * LDS_STRIDE];

  const int hd = blockIdx.y;                       // 0..17  (h*9 + d)
  const float* Whd = W + (size_t)hd * OUT_F * IN_F;

  // Stage W[hd] (64x256 f32 = 64KB) into LDS, padded rows.
  for (int i = threadIdx.x; i < OUT_F * IN_F; i += 256) {
    const int o = i >> 8;        // / IN_F
    const int k = i & (IN_F - 1);
    w_lds[o * LDS_STRIDE + k] = Whd[i];
  }
  __syncthreads();

  const int wave = threadIdx.x >> 5;
  const int lane = threadIdx.x & 31;
  const int half = lane >> 4;       // 0: K pair {0,1}, 1: K pair {2,3}
  const int l16  = lane & 15;

  const int tile = blockIdx.x * 8 + wave;          // 16-node tile index
  if (tile >= N_NODES / 16) return;                // wave-uniform (EXEC stays full)

  const int node = tile * 16 + l16;                // A row M = l16 (both halves)
  const float nv = norm[node];
  const float* frow = features + (size_t)node * IN_F;

  v8f acc[4];
#pragma unroll
  for (int t = 0; t < 4; ++t) acc[t] = (v8f){};

  for (int k0 = 0; k0 < IN_F; k0 += 4) {
    const int ka = k0 + half * 2;
    // A fragment (16x4 f32): lane l16 row, VGPR0=K ka, VGPR1=K ka+1
    v2f a;
    a.x = frow[ka]     * nv;
    a.y = frow[ka + 1] * nv;
#pragma unroll
    for (int t = 0; t < 4; ++t) {
      // B fragment (4x16 f32): col N = t*16 + l16, rows striped like A's K split
      const float* wrow = &w_lds[(t * 16 + l16) * LDS_STRIDE + ka];
      v2f b;
      b.x = wrow[0];
      b.y = wrow[1];
      acc[t] = __builtin_amdgcn_wmma_f32_16x16x4_f32(
          /*neg_a=*/false, a, /*neg_b=*/false, b,
          /*c_mod=*/(short)0, acc[t], /*reuse_a=*/false, /*reuse_b=*/false);
    }
  }

  // C/D layout: VGPR v, lanes 0-15 -> M=v, lanes 16-31 -> M=v+8; N=l16
  float* hbase = h_ws + ((size_t)hd * N_NODES + (size_t)tile * 16) * OUT_F;
#pragma unroll
  for (int t = 0; t < 4; ++t) {
#pragma unroll
    for (int v = 0; v < 8; ++v) {
      const int m = v + 8 * half;
      hbase[(size_t)m * OUT_F + t * 16 + l16] = acc[t][v];
    }
  }
}

// -----------------------------------------------------------------------------
// Kernel B: zero the output accumulator (d_out is poisoned by the harness).
// -----------------------------------------------------------------------------
__global__ __launch_bounds__(256) void zero_out(float* __restrict__ out, int n) {
  const int i = blockIdx.x * 256 + threadIdx.x;
  if (i < n) out[i] = 0.0f;
}

// -----------------------------------------------------------------------------
// Kernel C: per-edge scatter. 128 threads/edge: t = head*64 + o.
// out[dst, head*576 + rel*64 + o] += h_ws[(head*9+rel)][src][o]
// (segment_sum + all reference transposes fused into the atomic index)
// -----------------------------------------------------------------------------
__global__ __launch_bounds__(256) void edge_scatter(
    const float* __restrict__ h_ws,
    const int* __restrict__ edge_index,     // [2][E]: row0=dst, row1=src
    const int* __restrict__ edge_relation,  // [E]
    float* __restrict__ out) {
  const long long gid = (long long)blockIdx.x * 256 + threadIdx.x;
  const long long e = gid >> 7;
  if (e >= N_EDGES) return;
  const int t    = (int)(gid & 127);
  const int head = t >> 6;
  const int o    = t & 63;

  const int dst = edge_index[e];
  const int src = edge_index[N_EDGES + e];
  const int rel = edge_relation[e];

  const float val =
      h_ws[(((size_t)(head * NUM_DIV + rel)) * N_NODES + src) * OUT_F + o];
  atomicAdd(&out[(size_t)dst * OUT_COLS + head * (NUM_DIV * OUT_F) + rel * OUT_F + o],
            val);
}

// -----------------------------------------------------------------------------
// Kernel D: out = relu(out * norm[n]) in place.
// -----------------------------------------------------------------------------
__global__ __launch_bounds__(256) void finalize(
    float* __restrict__ out, const float* __restrict__ norm, int n) {
  const int i = blockIdx.x * 256 + threadIdx.x;
  if (i >= n) return;
  const int node = i / OUT_COLS;
  const float v = out[i] * norm[node];
  out[i] = v > 0.0f ? v : 0.0f;
}

// -----------------------------------------------------------------------------
extern "C" void kernel_launch(void* const* d_in, const int* in_sizes, int n_in,
                              void* d_out, int out_size, void* d_ws, size_t ws_size,
                              hipStream_t stream) {
  const float* features      = (const float*)d_in[0];  // [50000, 256]
  const float* norm          = (const float*)d_in[1];  // [50000, 1]
  const float* W             = (const float*)d_in[2];  // [2, 9, 64, 256]
  const int*   edge_index    = (const int*)d_in[3];    // [2, 800000]
  const int*   edge_relation = (const int*)d_in[4];    // [800000]
  float*       out           = (float*)d_out;          // [50000, 1152]
  float*       h_ws          = (float*)d_ws;           // 18*50000*64 f32 = 230.4 MB

  (void)in_sizes; (void)n_in; (void)out_size; (void)ws_size;

  const int out_elems = N_NODES * OUT_COLS;            // 57,600,000

  // 1) zero output accumulator
  zero_out<<<(out_elems + 255) / 256, 256, 0, stream>>>(out, out_elems);

  // 2) scaled GEMM -> h_ws  (grid: 3125 node-tiles / 8 waves, 18 hd slices)
  dim3 ggrid((N_NODES / 16 + 7) / 8, HD);
  geomgcn_gemm<<<ggrid, 256, 0, stream>>>(features, norm, W, h_ws);

  // 3) edge scatter with f32 atomics  (800000 edges * 128 threads)
  const long long sthreads = (long long)N_EDGES * 128;
  edge_scatter<<<(unsigned)((sthreads + 255) / 256), 256, 0, stream>>>(
      h_ws, edge_index, edge_relation, out);

  // 4) relu(out * norm) in place
  finalize<<<(out_elems + 255) / 256, 256, 0, stream>>>(out, norm, out_elems);
}